// NNCLR_14448269984034
// MI455X (gfx1250) — compile-verified
//
#include <hip/hip_runtime.h>
#include <hip/hip_bf16.h>

// ---------------------------------------------------------------------------
// NNCLR loss on MI455X (gfx1250), fp32 WMMA (V_WMMA_F32_16X16X4_F32).
// LDS-tiled NT GEMMs with register double-buffered staging for the two
// bandwidth-dominant stages (queue argmax + the four GEMM/logsumexp passes).
// ---------------------------------------------------------------------------

typedef __attribute__((ext_vector_type(2))) float v2f;
typedef __attribute__((ext_vector_type(8))) float v8f;

static constexpr int Bn = 4096;    // batch
static constexpr int Dd = 256;     // input feature dim
static constexpr int Pp = 256;     // projection dim
static constexpr int Qq = 32768;   // queue rows
static constexpr int BT = 8192;    // 2*Bn (both views stacked)
static constexpr float INV_T = 10.0f;  // 1/temperature
static constexpr float SHIFT = 10.0f;  // max possible logit (|cos|<=1, /t)
static constexpr int NCH_AM  = 8;   // argmax column chunks
static constexpr int NCH_LSE = 8;   // lse column chunks

// Tiled-GEMM geometry
static constexpr int BM = 64;    // block rows
static constexpr int BN = 128;   // block cols
static constexpr int KC = 64;    // K chunk staged in LDS
static constexpr int KP = 68;    // padded LDS row stride (68 % 64 == 4 -> no bank conflicts)
static constexpr int AV4 = (BM * KC / 4) / 256;  // float4 per thread for A chunk (= 4)
static constexpr int BV4 = (BN * KC / 4) / 256;  // float4 per thread for B chunk (= 8)

#define WMMA_F32(a, b, c) \
  __builtin_amdgcn_wmma_f32_16x16x4_f32(false, (a), false, (b), (short)0, (c), false, false)

// ---- simple per-wave tile (used by the small MLP GEMMs) -------------------
// C = A(16xK, row-major) @ W(KxN tile, row-major, ld=ldw)
__device__ __forceinline__ v8f wmma_tile_nn(const float* __restrict__ A,
                                            const float* __restrict__ W,
                                            int K, int ldw) {
  const int lane = threadIdx.x & 31;
  const int mn = lane & 15;
  const int kh = (lane >> 4) * 2;
  const float* ap = A + (size_t)mn * K + kh;
  const float* wp = W + (size_t)kh * ldw + mn;
  v8f c = {0.f, 0.f, 0.f, 0.f, 0.f, 0.f, 0.f, 0.f};
#pragma unroll 8
  for (int k = 0; k < K; k += 4) {
    v2f a = *reinterpret_cast<const v2f*>(ap + k);
    v2f b;
    b.x = wp[0];
    b.y = wp[ldw];
    c = WMMA_F32(a, b, c);
    wp += 4 * ldw;
  }
  return c;
}

// ---- chunk staging: global -> regs, regs -> LDS ---------------------------
template <int NV4>
__device__ __forceinline__ void load_chunk_regs(float4* r, const float* __restrict__ src,
                                                int gRowBase, int kc, int K, int tid) {
#pragma unroll
  for (int i = 0; i < NV4; ++i) {
    const int idx = tid + i * 256;
    const int rr = idx >> 4;      // KC/4 == 16 float4 per row
    const int j = idx & 15;
    r[i] = *reinterpret_cast<const float4*>(src + (size_t)(gRowBase + rr) * K + kc + j * 4);
  }
}

template <int NV4>
__device__ __forceinline__ void store_chunk_lds(float* dst, const float4* r, int tid) {
#pragma unroll
  for (int i = 0; i < NV4; ++i) {
    const int idx = tid + i * 256;
    const int rr = idx >> 4;
    const int j = idx & 15;
    *reinterpret_cast<float4*>(dst + rr * KP + j * 4) = r[i];
  }
}

// ---- Stage 1/2: GEMM + bias (+ optional relu), one wave per 16x16 tile ----
__global__ void k_gemm_bias(const float* __restrict__ In,
                            const float* __restrict__ W,
                            const float* __restrict__ bias,
                            float* __restrict__ Out,
                            int K, int N, int doRelu) {
  const int rowBase = blockIdx.x * 16;
  const int colBase = blockIdx.y * 16;
  v8f c = wmma_tile_nn(In + (size_t)rowBase * K, W + colBase, K, N);
  const int lane = threadIdx.x & 31;
  const int n = colBase + (lane & 15);
  const int half = lane >> 4;
  const float bv = bias[n];
#pragma unroll
  for (int v = 0; v < 8; ++v) {
    float x = c[v] + bv;
    if (doRelu) x = fmaxf(x, 0.f);
    Out[(size_t)(rowBase + v + 8 * half) * N + n] = x;
  }
}

// ---- Stage 2b: row L2 normalize (in place), N == 256 ----------------------
__global__ void k_rownorm(float* __restrict__ P, int N) {
  const int row = blockIdx.x;
  const int lane = threadIdx.x & 31;
  float* r = P + (size_t)row * N;
  float x[8];
  float ss = 0.f;
#pragma unroll
  for (int j = 0; j < 8; ++j) { x[j] = r[lane * 8 + j]; ss += x[j] * x[j]; }
#pragma unroll
  for (int off = 16; off > 0; off >>= 1) ss += __shfl_xor(ss, off, 32);
  const float s = 1.0f / fmaxf(sqrtf(ss), 1e-12f);
#pragma unroll
  for (int j = 0; j < 8; ++j) r[lane * 8 + j] = x[j] * s;
}

// ---- Stage 3: LDS-tiled cosine-sim argmax vs queue ------------------------
// grid (BT/BM, NCH_AM), block 256 (8 waves: 2 row-groups x 4 col-groups)
__global__ __launch_bounds__(256) void k_argmax_tiled(
    const float* __restrict__ P, const float* __restrict__ Qm,
    float* __restrict__ pval, int* __restrict__ pidx,
    int K, int colsPerChunk, int nch) {
  __shared__ float As[BM * KP];
  __shared__ float Bs[BN * KP];
  __shared__ float candV[BM][4];
  __shared__ int candI[BM][4];

  const int tid = threadIdx.x;
  const int wave = tid >> 5;
  const int lane = tid & 31;
  const int mn = lane & 15;
  const int kh = (lane >> 4) * 2;
  const int half = lane >> 4;
  const int wr = wave & 1;   // row group of 32
  const int wc = wave >> 1;  // col group of 32 (0..3)
  const int rowBase = blockIdx.x * BM;
  const int colStart = blockIdx.y * colsPerChunk;

  float best[2][8];
  int bidx[2][8];
#pragma unroll
  for (int rt = 0; rt < 2; ++rt)
#pragma unroll
    for (int v = 0; v < 8; ++v) { best[rt][v] = -3.4e38f; bidx[rt][v] = 0x7fffffff; }

  float4 aR[AV4], bR[BV4];
  load_chunk_regs<AV4>(aR, P, rowBase, 0, K, tid);
  load_chunk_regs<BV4>(bR, Qm, colStart, 0, K, tid);

  for (int cb = 0; cb < colsPerChunk; cb += BN) {
    v8f c00 = {0.f,0.f,0.f,0.f,0.f,0.f,0.f,0.f};
    v8f c01 = c00, c10 = c00, c11 = c00;
    for (int kc = 0; kc < K; kc += KC) {
      __syncthreads();                       // previous chunk's compute done
      store_chunk_lds<AV4>(As, aR, tid);
      store_chunk_lds<BV4>(Bs, bR, tid);
      __syncthreads();
      // issue next chunk's global loads; they overlap the WMMA phase below
      int nKc = kc + KC, nCb = cb;
      if (nKc == K) { nKc = 0; nCb = cb + BN; }
      if (nCb < colsPerChunk) {
        load_chunk_regs<AV4>(aR, P, rowBase, nKc, K, tid);
        load_chunk_regs<BV4>(bR, Qm, colStart + nCb, nKc, K, tid);
      }
      const float* ap0 = As + (wr * 32 + mn) * KP + kh;
      const float* ap1 = ap0 + 16 * KP;
      const float* bp0 = Bs + (wc * 32 + mn) * KP + kh;
      const float* bp1 = bp0 + 16 * KP;
#pragma unroll
      for (int k = 0; k < KC; k += 4) {
        v2f a0 = *reinterpret_cast<const v2f*>(ap0 + k);
        v2f a1 = *reinterpret_cast<const v2f*>(ap1 + k);
        v2f b0 = *reinterpret_cast<const v2f*>(bp0 + k);
        v2f b1 = *reinterpret_cast<const v2f*>(bp1 + k);
        c00 = WMMA_F32(a0, b0, c00);
        c01 = WMMA_F32(a0, b1, c01);
        c10 = WMMA_F32(a1, b0, c10);
        c11 = WMMA_F32(a1, b1, c11);
      }
    }
    // running argmax update (strict > in ascending col order -> first-max wins)
    const int colA = colStart + cb + wc * 32 + mn;
    const int colB = colA + 16;
#pragma unroll
    for (int v = 0; v < 8; ++v) {
      if (c00[v] > best[0][v]) { best[0][v] = c00[v]; bidx[0][v] = colA; }
      if (c01[v] > best[0][v]) { best[0][v] = c01[v]; bidx[0][v] = colB; }
      if (c10[v] > best[1][v]) { best[1][v] = c10[v]; bidx[1][v] = colA; }
      if (c11[v] > best[1][v]) { best[1][v] = c11[v]; bidx[1][v] = colB; }
    }
  }

  // merge across the 16 lanes of each half (tie-break: smaller index)
#pragma unroll
  for (int rt = 0; rt < 2; ++rt)
#pragma unroll
    for (int v = 0; v < 8; ++v)
#pragma unroll
      for (int off = 1; off < 16; off <<= 1) {
        const float ov = __shfl_xor(best[rt][v], off, 32);
        const int oi = __shfl_xor(bidx[rt][v], off, 32);
        if (ov > best[rt][v] || (ov == best[rt][v] && oi < bidx[rt][v])) {
          best[rt][v] = ov;
          bidx[rt][v] = oi;
        }
      }
  if (mn == 0) {
#pragma unroll
    for (int rt = 0; rt < 2; ++rt)
#pragma unroll
      for (int v = 0; v < 8; ++v) {
        const int rl = wr * 32 + rt * 16 + v + 8 * half;
        candV[rl][wc] = best[rt][v];
        candI[rl][wc] = bidx[rt][v];
      }
  }
  __syncthreads();
  if (tid < BM) {
    float bv = candV[tid][0];
    int bi = candI[tid][0];
#pragma unroll
    for (int w = 1; w < 4; ++w) {
      const float ov = candV[tid][w];
      const int oi = candI[tid][w];
      if (ov > bv || (ov == bv && oi < bi)) { bv = ov; bi = oi; }
    }
    const int row = rowBase + tid;
    pval[(size_t)row * nch + blockIdx.y] = bv;
    pidx[(size_t)row * nch + blockIdx.y] = bi;
  }
}

// ---- Stage 3b: finalize argmax over chunks + gather queue row -------------
__global__ void k_argmax_fin(const float* __restrict__ pval,
                             const int* __restrict__ pidx,
                             const float* __restrict__ Qm,
                             float* __restrict__ NN, int nch, int K) {
  const int row = blockIdx.x;
  const int lane = threadIdx.x & 31;
  float bv = -3.4e38f;
  int bi = 0x7fffffff;
  if (lane < nch) {
    bv = pval[(size_t)row * nch + lane];
    bi = pidx[(size_t)row * nch + lane];
  }
#pragma unroll
  for (int off = 1; off < 32; off <<= 1) {
    const float ov = __shfl_xor(bv, off, 32);
    const int oi = __shfl_xor(bi, off, 32);
    if (ov > bv || (ov == bv && oi < bi)) { bv = ov; bi = oi; }
  }
  const float* src = Qm + (size_t)bi * K;
  float* dst = NN + (size_t)row * K;
  for (int j = lane; j < K; j += 32) dst[j] = src[j];
}

// ---- Stage 4: LDS-tiled GEMM (A @ Bt^T) + row sum(exp(s/t - SHIFT)) -------
// grid (Bn/BM, NCH_LSE), block 256. Optionally captures the diagonal s/t.
__global__ __launch_bounds__(256) void k_lse_tiled(
    const float* __restrict__ A, const float* __restrict__ Bt,
    float* __restrict__ pe, float* __restrict__ diag,
    int K, int colsPerChunk, int nch, int writeDiag) {
  __shared__ float As[BM * KP];
  __shared__ float Bs[BN * KP];
  __shared__ float candV[BM][4];

  const int tid = threadIdx.x;
  const int wave = tid >> 5;
  const int lane = tid & 31;
  const int mn = lane & 15;
  const int kh = (lane >> 4) * 2;
  const int half = lane >> 4;
  const int wr = wave & 1;
  const int wc = wave >> 1;
  const int rowBase = blockIdx.x * BM;
  const int colStart = blockIdx.y * colsPerChunk;

  float acc[2][8];
#pragma unroll
  for (int rt = 0; rt < 2; ++rt)
#pragma unroll
    for (int v = 0; v < 8; ++v) acc[rt][v] = 0.f;

  float4 aR[AV4], bR[BV4];
  load_chunk_regs<AV4>(aR, A, rowBase, 0, K, tid);
  load_chunk_regs<BV4>(bR, Bt, colStart, 0, K, tid);

  for (int cb = 0; cb < colsPerChunk; cb += BN) {
    v8f c00 = {0.f,0.f,0.f,0.f,0.f,0.f,0.f,0.f};
    v8f c01 = c00, c10 = c00, c11 = c00;
    for (int kc = 0; kc < K; kc += KC) {
      __syncthreads();
      store_chunk_lds<AV4>(As, aR, tid);
      store_chunk_lds<BV4>(Bs, bR, tid);
      __syncthreads();
      int nKc = kc + KC, nCb = cb;
      if (nKc == K) { nKc = 0; nCb = cb + BN; }
      if (nCb < colsPerChunk) {
        load_chunk_regs<AV4>(aR, A, rowBase, nKc, K, tid);
        load_chunk_regs<BV4>(bR, Bt, colStart + nCb, nKc, K, tid);
      }
      const float* ap0 = As + (wr * 32 + mn) * KP + kh;
      const float* ap1 = ap0 + 16 * KP;
      const float* bp0 = Bs + (wc * 32 + mn) * KP + kh;
      const float* bp1 = bp0 + 16 * KP;
#pragma unroll
      for (int k = 0; k < KC; k += 4) {
        v2f a0 = *reinterpret_cast<const v2f*>(ap0 + k);
        v2f a1 = *reinterpret_cast<const v2f*>(ap1 + k);
        v2f b0 = *reinterpret_cast<const v2f*>(bp0 + k);
        v2f b1 = *reinterpret_cast<const v2f*>(bp1 + k);
        c00 = WMMA_F32(a0, b0, c00);
        c01 = WMMA_F32(a0, b1, c01);
        c10 = WMMA_F32(a1, b0, c10);
        c11 = WMMA_F32(a1, b1, c11);
      }
    }
    const int colA = colStart + cb + wc * 32 + mn;
    const int colB = colA + 16;
#pragma unroll
    for (int v = 0; v < 8; ++v) {
      const int r0 = rowBase + wr * 32 + v + 8 * half;
      const int r1 = r0 + 16;
      const float s00 = c00[v] * INV_T, s01 = c01[v] * INV_T;
      const float s10 = c10[v] * INV_T, s11 = c11[v] * INV_T;
      acc[0][v] += __expf(s00 - SHIFT) + __expf(s01 - SHIFT);
      acc[1][v] += __expf(s10 - SHIFT) + __expf(s11 - SHIFT);
      if (writeDiag) {
        if (colA == r0) diag[r0] = s00;
        if (colB == r0) diag[r0] = s01;
        if (colA == r1) diag[r1] = s10;
        if (colB == r1) diag[r1] = s11;
      }
    }
  }

#pragma unroll
  for (int rt = 0; rt < 2; ++rt)
#pragma unroll
    for (int v = 0; v < 8; ++v)
#pragma unroll
      for (int off = 1; off < 16; off <<= 1)
        acc[rt][v] += __shfl_xor(acc[rt][v], off, 32);
  if (mn == 0) {
#pragma unroll
    for (int rt = 0; rt < 2; ++rt)
#pragma unroll
      for (int v = 0; v < 8; ++v) {
        const int rl = wr * 32 + rt * 16 + v + 8 * half;
        candV[rl][wc] = acc[rt][v];
      }
  }
  __syncthreads();
  if (tid < BM) {
    const float s = candV[tid][0] + candV[tid][1] + candV[tid][2] + candV[tid][3];
    pe[(size_t)(rowBase + tid) * nch + blockIdx.y] = s;
  }
}

// ---- Stage 5: combine lse partials + diagonals into the scalar loss -------
__global__ void k_loss(const float* __restrict__ pe,  // [4][Bn][nch]
                       const float* __restrict__ d1,
                       const float* __restrict__ d2,
                       float* __restrict__ out, int nch) {
  __shared__ float sdata[256];
  const int tid = threadIdx.x;
  float local = 0.f;
  for (int row = tid; row < Bn; row += 256) {
    float lsum = 0.f;
#pragma unroll
    for (int pass = 0; pass < 4; ++pass) {
      const float* p = pe + ((size_t)pass * Bn + row) * nch;
      float s = 0.f;
      for (int ch = 0; ch < nch; ++ch) s += p[ch];
      lsum += SHIFT + __logf(s);
    }
    local += 2.f * d1[row] + 2.f * d2[row] - lsum;
  }
  sdata[tid] = local;
  __syncthreads();
  for (int off = 128; off > 0; off >>= 1) {
    if (tid < off) sdata[tid] += sdata[tid + off];
    __syncthreads();
  }
  if (tid == 0) out[0] = -sdata[0] / (4.0f * (float)Bn);
}

// ---------------------------------------------------------------------------
extern "C" void kernel_launch(void* const* d_in, const int* in_sizes, int n_in,
                              void* d_out, int out_size, void* d_ws,
                              size_t ws_size, hipStream_t stream) {
  (void)in_sizes; (void)n_in; (void)out_size; (void)ws_size;
  const float* f1 = (const float*)d_in[0];   // [4096,256]
  const float* f2 = (const float*)d_in[1];   // [4096,256]
  const float* W1 = (const float*)d_in[2];   // [256,256]
  const float* b1 = (const float*)d_in[3];   // [256]
  const float* W2 = (const float*)d_in[4];   // [256,256]
  const float* b2 = (const float*)d_in[5];   // [256]
  const float* Qm = (const float*)d_in[6];   // [32768,256]
  float* out = (float*)d_out;

  // Workspace layout (~24 MiB): H region is reused for small partials once dead.
  char* ws = (char*)d_ws;
  const size_t MB = 1024ull * 1024ull;
  float* P  = (float*)(ws);             // [8192,256]  projections (normed)
  float* NN = (float*)(ws + 8 * MB);    // [8192,256]  nearest-neighbour rows
  float* H  = (float*)(ws + 16 * MB);   // [8192,256]  hidden (dead after GEMM2)
  float* pval = (float*)(ws + 16 * MB);                 // [8192,8]   (reuses H)
  int*   pidx = (int*)  (ws + 16 * MB + 256 * 1024);    // [8192,8]
  float* pe   = (float*)(ws + 17 * MB);                 // [4][4096][8]
  float* dg1  = (float*)(ws + 17 * MB + 512 * 1024);    // [4096]
  float* dg2  = (float*)(ws + 17 * MB + 512 * 1024 + 16 * 1024);  // [4096]

  const dim3 wv(32);
  const dim3 blk(256);

  // Stage 1: H = relu(F @ W1 + b1), both views stacked
  k_gemm_bias<<<dim3(Bn / 16, Pp / 16), wv, 0, stream>>>(f1, W1, b1, H, Dd, Pp, 1);
  k_gemm_bias<<<dim3(Bn / 16, Pp / 16), wv, 0, stream>>>(
      f2, W1, b1, H + (size_t)Bn * Pp, Dd, Pp, 1);

  // Stage 2: P = H @ W2 + b2, then row-wise L2 normalize
  k_gemm_bias<<<dim3(BT / 16, Pp / 16), wv, 0, stream>>>(H, W2, b2, P, Pp, Pp, 0);
  k_rownorm<<<dim3(BT), wv, 0, stream>>>(P, Pp);

  // Stage 3: argmax over queue (tiled, split over column chunks) + gather
  k_argmax_tiled<<<dim3(BT / BM, NCH_AM), blk, 0, stream>>>(
      P, Qm, pval, pidx, Pp, Qq / NCH_AM, NCH_AM);
  k_argmax_fin<<<dim3(BT), wv, 0, stream>>>(pval, pidx, Qm, NN, NCH_AM, Pp);

  // Stage 4: four fused GEMM + row-logsumexp passes (col-lse == row-lse of S^T)
  const float* p1 = P;
  const float* p2 = P + (size_t)Bn * Pp;
  const float* nn1 = NN;
  const float* nn2 = NN + (size_t)Bn * Pp;
  const size_t passStride = (size_t)Bn * NCH_LSE;
  const int cpc = Bn / NCH_LSE;
  k_lse_tiled<<<dim3(Bn / BM, NCH_LSE), blk, 0, stream>>>(
      nn1, p2, pe + 0 * passStride, dg1, Pp, cpc, NCH_LSE, 1);  // rows of S1
  k_lse_tiled<<<dim3(Bn / BM, NCH_LSE), blk, 0, stream>>>(
      p2, nn1, pe + 1 * passStride, dg1, Pp, cpc, NCH_LSE, 0);  // rows of S1^T
  k_lse_tiled<<<dim3(Bn / BM, NCH_LSE), blk, 0, stream>>>(
      nn2, p1, pe + 2 * passStride, dg2, Pp, cpc, NCH_LSE, 1);  // rows of S2
  k_lse_tiled<<<dim3(Bn / BM, NCH_LSE), blk, 0, stream>>>(
      p1, nn2, pe + 3 * passStride, dg2, Pp, cpc, NCH_LSE, 0);  // rows of S2^T

  // Stage 5: scalar loss
  k_loss<<<dim3(1), dim3(256), 0, stream>>>(pe, dg1, dg2, out, NCH_LSE);
}